// GCN_N_3530463118086
// MI455X (gfx1250) — compile-verified
//
#include <hip/hip_runtime.h>

typedef __attribute__((ext_vector_type(2))) float v2f;
typedef __attribute__((ext_vector_type(8))) float v8f;

#define DIM 64

// ---------------- elementwise / setup kernels ----------------

__global__ void fill_kernel(float* __restrict__ p, float v, int n) {
  int i = blockIdx.x * blockDim.x + threadIdx.x;
  if (i < n) p[i] = v;
}

__global__ void deg_kernel(const int* __restrict__ dst, float* __restrict__ deg, int E) {
  int e = blockIdx.x * blockDim.x + threadIdx.x;
  if (e < E) atomicAdd(&deg[dst[e]], 1.0f);
}

__global__ void rsqrt_kernel(float* __restrict__ d, int n) {
  int i = blockIdx.x * blockDim.x + threadIdx.x;
  if (i < n) {
    float v = d[i];
    d[i] = (v > 0.0f) ? rsqrtf(v) : 0.0f;
  }
}

// ---------------- WMMA GEMM: Y[nrows x 64] = X[nrows x K] * W[K x 64] ----------------
// V_WMMA_F32_16X16X4_F32; one wave per 16-row tile; 4 N-tiles of 16 cols.
// A frag (16x4 f32): lane L<16: row=L,    {K=k0,   K=k0+1}
//                    lane L>=16: row=L-16, {K=k0+2, K=k0+3}
// B frag (4x16 f32): lane L<16: col=L,    {row k0,   row k0+1}
//                    lane L>=16: col=L-16, {row k0+2, row k0+3}
// LDS W is K-pair interleaved: sW[(k>>1)*128 + col*2 + (k&1)] = W[k][col]
// so each B fragment is a single aligned ds_load_b64.
// C/D: VGPR v -> M = v (lanes 0-15) / v+8 (lanes 16-31), N = lane%16.
__global__ void __launch_bounds__(256) gemm_wmma(const float* __restrict__ X,
                                                 const float* __restrict__ W,
                                                 float* __restrict__ Y,
                                                 int nrows, int K) {
  __shared__ float sW[128 * DIM];  // up to K=128 (32 KB)
  const int tid = threadIdx.x;
  for (int i = tid; i < K * DIM; i += 256) {
    int k = i >> 6;          // i / DIM
    int col = i & 63;        // i % DIM
    sW[(k >> 1) * 128 + col * 2 + (k & 1)] = W[i];
  }
  __syncthreads();

  const int wave = tid >> 5;
  const int lane = tid & 31;
  const int tile = blockIdx.x * 8 + wave;
  const int row0 = tile * 16;
  if (row0 >= nrows) return;  // wave-uniform: EXEC stays all-1s for WMMA

  const int m = lane & 15;
  const int khalf = (lane >> 4) << 1;  // 0 or 2

  int rowA = row0 + m;
  if (rowA >= nrows) rowA = nrows - 1;  // clamp loads; stores guarded below
  const float* __restrict__ xr = X + (size_t)rowA * K;

  v8f acc0 = {}, acc1 = {}, acc2 = {}, acc3 = {};

  for (int k0 = 0; k0 < K; k0 += 4) {
    const int kk = k0 + khalf;         // even
    v2f a = *(const v2f*)&xr[kk];      // global_load_b64

    const float* wp = &sW[(kk >> 1) * 128 + m * 2];
    v2f b0 = *(const v2f*)&wp[0];      // ds_load_b64 each
    v2f b1 = *(const v2f*)&wp[32];
    v2f b2 = *(const v2f*)&wp[64];
    v2f b3 = *(const v2f*)&wp[96];

    acc0 = __builtin_amdgcn_wmma_f32_16x16x4_f32(false, a, false, b0, (short)0, acc0, false, false);
    acc1 = __builtin_amdgcn_wmma_f32_16x16x4_f32(false, a, false, b1, (short)0, acc1, false, false);
    acc2 = __builtin_amdgcn_wmma_f32_16x16x4_f32(false, a, false, b2, (short)0, acc2, false, false);
    acc3 = __builtin_amdgcn_wmma_f32_16x16x4_f32(false, a, false, b3, (short)0, acc3, false, false);
  }

  const int rOff = (lane >> 4) << 3;  // 0 or 8
#pragma unroll
  for (int v = 0; v < 8; ++v) {
    int row = row0 + rOff + v;
    if (row < nrows) {
      float* yr = Y + (size_t)row * DIM + m;
      yr[0]  = acc0[v];
      yr[16] = acc1[v];
      yr[32] = acc2[v];
      yr[48] = acc3[v];
    }
  }
}

// ------- edge scatter: agg[dst] += dinv[src]*dinv[dst] * m[src], 2 feats/thread -------
__global__ void scatter_kernel(const int* __restrict__ src, const int* __restrict__ dst,
                               const float* __restrict__ dinv, const float* __restrict__ m,
                               float* __restrict__ agg, int E) {
  int t = blockIdx.x * blockDim.x + threadIdx.x;
  if (t >= E * (DIM / 2)) return;
  int e = t >> 5;            // edge
  int f = (t & 31) << 1;     // feature pair base
  int s = src[e];
  int d = dst[e];
  float c = dinv[s] * dinv[d];
  v2f mv = *(const v2f*)&m[(size_t)s * DIM + f];
  float* ap = &agg[(size_t)d * DIM + f];
  atomicAdd(&ap[0], c * mv.x);
  atomicAdd(&ap[1], c * mv.y);
}

// -------- h = relu(agg + 2*dinv^2*m + b), written in place over agg --------
__global__ void bias_relu_self(float* __restrict__ agg, const float* __restrict__ m,
                               const float* __restrict__ dinv, const float* __restrict__ b,
                               int N) {
  int t = blockIdx.x * blockDim.x + threadIdx.x;
  if (t >= N * DIM) return;
  int n = t >> 6;
  int f = t & 63;
  float di = dinv[n];
  float v = agg[t] + 2.0f * di * di * m[t] + b[f];
  agg[t] = (v > 0.0f) ? v : 0.0f;
}

// ---------------- global add pool ----------------
__global__ void pool_kernel(const float* __restrict__ h, const int* __restrict__ batch,
                            float* __restrict__ pooled, int N) {
  int t = blockIdx.x * blockDim.x + threadIdx.x;
  if (t >= N * DIM) return;
  int n = t >> 6;
  int f = t & 63;
  atomicAdd(&pooled[(size_t)batch[n] * DIM + f], h[t]);
}

// ---------------- final projection: out[g] = pooled[g,:] . Wo + bo ----------------
__global__ void out_kernel(const float* __restrict__ pooled, const float* __restrict__ Wo,
                           const float* __restrict__ bo, float* __restrict__ out, int G) {
  int g = blockIdx.x * blockDim.x + threadIdx.x;
  if (g >= G) return;
  float s = bo[0];
#pragma unroll
  for (int f = 0; f < DIM; ++f) s += pooled[(size_t)g * DIM + f] * Wo[f];
  out[g] = s;
}

// ---------------- host-side orchestration ----------------

static inline int cdiv(long long a, long long b) { return (int)((a + b - 1) / b); }

extern "C" void kernel_launch(void* const* d_in, const int* in_sizes, int n_in,
                              void* d_out, int out_size, void* d_ws, size_t ws_size,
                              hipStream_t stream) {
  const float* x     = (const float*)d_in[0];
  const int*   edge  = (const int*)d_in[1];
  const int*   batch = (const int*)d_in[2];
  const float* W1 = (const float*)d_in[3];
  const float* b1 = (const float*)d_in[4];
  const float* Wn = (const float*)d_in[5];
  const float* bn = (const float*)d_in[6];
  const float* Wo = (const float*)d_in[7];
  const float* bo = (const float*)d_in[8];
  float* out = (float*)d_out;

  const int N    = in_sizes[2];          // batch vector length
  const int E    = in_sizes[1] / 2;      // edge_index is [2, E]
  const int F_IN = in_sizes[0] / N;      // 128
  const int G    = out_size;             // 64

  const int* src = edge;       // edge_index[0]
  const int* dst = edge + E;   // edge_index[1]

  const size_t nd = (size_t)N * DIM;
  float* ws     = (float*)d_ws;
  float* bufQ   = ws;             // m (messages), reused every layer
  float* bufA   = bufQ + nd;      // agg/h ping
  float* bufB   = bufA + nd;      // agg/h pong
  float* dinv   = bufB + nd;      // degree -> dinv, N floats
  float* pooled = dinv + N;       // G*DIM floats

  const int nd_i = (int)nd;
  const int gemmBlocks = cdiv(cdiv(N, 16), 8);
  const int edBlocks = cdiv((long long)E * (DIM / 2), 256);
  const int ndBlocks = cdiv(nd_i, 256);

  // degrees (self-loop weight 2.0 as fill value), then dinv = rsqrt(deg)
  fill_kernel<<<cdiv(N, 256), 256, 0, stream>>>(dinv, 2.0f, N);
  deg_kernel<<<cdiv(E, 256), 256, 0, stream>>>(dst, dinv, E);
  rsqrt_kernel<<<cdiv(N, 256), 256, 0, stream>>>(dinv, N);

  // ---- layer 1: x -> bufA ----
  gemm_wmma<<<gemmBlocks, 256, 0, stream>>>(x, W1, bufQ, N, F_IN);
  fill_kernel<<<ndBlocks, 256, 0, stream>>>(bufA, 0.0f, nd_i);
  scatter_kernel<<<edBlocks, 256, 0, stream>>>(src, dst, dinv, bufQ, bufA, E);
  bias_relu_self<<<ndBlocks, 256, 0, stream>>>(bufA, bufQ, dinv, b1, N);

  // ---- layer 2: bufA -> bufB ----
  gemm_wmma<<<gemmBlocks, 256, 0, stream>>>(bufA, Wn, bufQ, N, DIM);
  fill_kernel<<<ndBlocks, 256, 0, stream>>>(bufB, 0.0f, nd_i);
  scatter_kernel<<<edBlocks, 256, 0, stream>>>(src, dst, dinv, bufQ, bufB, E);
  bias_relu_self<<<ndBlocks, 256, 0, stream>>>(bufB, bufQ, dinv, bn, N);

  // ---- layer 3: bufB -> bufA ----
  gemm_wmma<<<gemmBlocks, 256, 0, stream>>>(bufB, Wn, bufQ, N, DIM);
  fill_kernel<<<ndBlocks, 256, 0, stream>>>(bufA, 0.0f, nd_i);
  scatter_kernel<<<edBlocks, 256, 0, stream>>>(src, dst, dinv, bufQ, bufA, E);
  bias_relu_self<<<ndBlocks, 256, 0, stream>>>(bufA, bufQ, dinv, bn, N);

  // ---- pool + project ----
  fill_kernel<<<cdiv(G * DIM, 256), 256, 0, stream>>>(pooled, 0.0f, G * DIM);
  pool_kernel<<<ndBlocks, 256, 0, stream>>>(bufA, batch, pooled, N);
  out_kernel<<<1, 64, 0, stream>>>(pooled, Wo, bo, out, G);
}